// GraphConvolution_2783138808134
// MI455X (gfx1250) — compile-verified
//
#include <hip/hip_runtime.h>
#include <hip/hip_bf16.h>
#include <stdint.h>

typedef __attribute__((ext_vector_type(2))) float v2f;
typedef __attribute__((ext_vector_type(4))) float v4f;
typedef __attribute__((ext_vector_type(8))) float v8f;

#define IN_CH  512
#define OUT_CH 512
#define KC     64          // K-chunk staged in LDS
#define XPAD   68          // sX row stride in floats (even -> 8B-aligned v2f reads)
#define WP2    80          // sW2 row stride in float2 elems; 160 dwords % 64 banks = 32
                           // -> K-pair rows kp / kp+1 use disjoint bank halves

// Async copy 16B memory -> LDS (GVS mode: SGPR64 base + VGPR i32 byte offset).
// Tracked by ASYNCcnt; per-lane under EXEC. No VGPR data return.
__device__ __forceinline__ void async_b128_to_lds(uint32_t lds_off,
                                                  const float* base,
                                                  uint32_t byte_off)
{
    asm volatile("global_load_async_to_lds_b128 %0, %1, %2"
                 :
                 : "v"(lds_off), "v"(byte_off), "s"(base)
                 : "memory");
}

__device__ __forceinline__ void wait_async_zero()
{
    asm volatile("s_wait_asynccnt 0" ::: "memory");
}

// ---------------------------------------------------------------------------
// Kernel 1: support = X @ W   (fp32, V_WMMA_F32_16X16X4_F32)
// Block = 128 threads (4 waves). Block tile = 128(M) x 64(N).
// Each wave owns a 32x64 strip: 2 M-tiles x 4 N-tiles = 8 accumulators, so
// every B fragment feeds 2 WMMAs (0.375 ds-ops/wmma in steady state).
// X staged via GLOBAL_LOAD_ASYNC_TO_LDS_B128 (no VGPR bounce);
// W staged K-paired so each B fragment is ONE aligned ds_load_b64.
// ---------------------------------------------------------------------------
__global__ __launch_bounds__(128) void gcn_gemm_wmma_f32(
    const float* __restrict__ X, const float* __restrict__ W,
    float* __restrict__ S, int M)
{
    __shared__ __align__(16) float sX[128 * XPAD];      // 34.0 KB
    __shared__ __align__(16) v2f   sW2[(KC / 2) * WP2]; // 20.0 KB

    const int tid  = threadIdx.x;
    const int lane = tid & 31;
    const int wave = tid >> 5;          // 0..3
    const int m0   = blockIdx.x * 128;
    const int n0   = blockIdx.y * 64;
    const int m_l  = lane & 15;         // M (for A) / N (for B,D) within tile
    const int kh   = lane >> 4;         // which K pair-half / M-half

    v8f acc0[4] = {};                   // M-tile 0 (rows m_base .. m_base+15)
    v8f acc1[4] = {};                   // M-tile 1 (rows m_base+16 .. m_base+31)
    const int m_base = m0 + wave * 32;

    for (int kc = 0; kc < IN_CH; kc += KC) {
        __syncthreads();
        // stage X chunk asynchronously: 128 rows x 64 k-cols, 16B per lane-op.
        // Rows >= M are skipped; they are never consumed (per-tile guards).
        {
            const int c4 = (tid & 15) * 4;
            const int r0 = tid >> 4;           // 0..7
            #pragma unroll
            for (int rp = 0; rp < 128; rp += 8) {
                const int r  = r0 + rp;
                const int gm = m0 + r;
                if (gm < M) {
                    const uint32_t lds_off =
                        (uint32_t)(uintptr_t)(&sX[r * XPAD + c4]);
                    const uint32_t goff =
                        (uint32_t)(((size_t)gm * IN_CH + kc + c4) * sizeof(float));
                    async_b128_to_lds(lds_off, X, goff);
                }
            }
        }
        // stage W chunk K-paired: sW2[kp][n] = {W[kc+2kp][n0+n], W[kc+2kp+1][n0+n]}
        {
            const int c4  = (tid & 15) * 4;    // n within 64 (4 consecutive)
            const int kp0 = tid >> 4;          // 0..7
            #pragma unroll
            for (int pp = 0; pp < 4; ++pp) {
                const int kp = kp0 + pp * 8;   // pair-row 0..31
                const float4 r0 = *(const float4*)(W + (size_t)(kc + 2 * kp    ) * OUT_CH + n0 + c4);
                const float4 r1 = *(const float4*)(W + (size_t)(kc + 2 * kp + 1) * OUT_CH + n0 + c4);
                v4f p0, p1;
                p0.x = r0.x; p0.y = r1.x; p0.z = r0.y; p0.w = r1.y;
                p1.x = r0.z; p1.y = r1.z; p1.z = r0.w; p1.w = r1.w;
                *(v4f*)&sW2[kp * WP2 + c4]     = p0;   // 16B-aligned ds_store_b128
                *(v4f*)&sW2[kp * WP2 + c4 + 2] = p1;
            }
        }
        wait_async_zero();      // our async LDS writes are done
        __syncthreads();        // publish staged tiles to all waves

        if (m_base < M) {
            const bool two = (m_base + 16 < M);           // wave-uniform
            const float* ax0 = &sX[(wave * 32 + m_l) * XPAD];
            const float* ax1 = ax0 + 16 * XPAD;
            if (two) {
                #pragma unroll
                for (int kk = 0; kk < KC; kk += 4) {
                    const int ka = kk + 2 * kh;
                    const v2f a0 = *(const v2f*)(ax0 + ka);
                    const v2f a1 = *(const v2f*)(ax1 + ka);
                    const int bidx = ((kk >> 1) + kh) * WP2 + m_l;
                    #pragma unroll
                    for (int t = 0; t < 4; ++t) {
                        const v2f b = sW2[bidx + t * 16];  // shared by both M-tiles
                        acc0[t] = __builtin_amdgcn_wmma_f32_16x16x4_f32(
                            false, a0, false, b, (short)0, acc0[t], false, false);
                        acc1[t] = __builtin_amdgcn_wmma_f32_16x16x4_f32(
                            false, a1, false, b, (short)0, acc1[t], false, false);
                    }
                }
            } else {
                #pragma unroll
                for (int kk = 0; kk < KC; kk += 4) {
                    const int ka = kk + 2 * kh;
                    const v2f a0 = *(const v2f*)(ax0 + ka);
                    const int bidx = ((kk >> 1) + kh) * WP2 + m_l;
                    #pragma unroll
                    for (int t = 0; t < 4; ++t) {
                        const v2f b = sW2[bidx + t * 16];
                        acc0[t] = __builtin_amdgcn_wmma_f32_16x16x4_f32(
                            false, a0, false, b, (short)0, acc0[t], false, false);
                    }
                }
            }
        }
    }

    if (m_base < M) {
        #pragma unroll
        for (int t = 0; t < 4; ++t) {
            #pragma unroll
            for (int i = 0; i < 8; ++i) {
                const int gm = m_base + i + 8 * kh;   // C/D layout: M = vgpr + 8*half
                S[(size_t)gm * OUT_CH + n0 + t * 16 + m_l] = acc0[t][i];
            }
        }
        if (m_base + 16 < M) {
            #pragma unroll
            for (int t = 0; t < 4; ++t) {
                #pragma unroll
                for (int i = 0; i < 8; ++i) {
                    const int gm = m_base + 16 + i + 8 * kh;
                    S[(size_t)gm * OUT_CH + n0 + t * 16 + m_l] = acc1[t][i];
                }
            }
        }
    }
}

// ---------------------------------------------------------------------------
// Kernel 2: COO scatter: out[row] += val * support[col]
// One 256-thread block per edge; float2 per thread covers 512 channels.
// support (102 MB) is L2-resident on MI455X (192 MB L2); atomics hit L2.
// ---------------------------------------------------------------------------
__global__ __launch_bounds__(256) void gcn_spmm_scatter(
    const float* __restrict__ S, const float* __restrict__ vals,
    const int* __restrict__ rows, const int* __restrict__ cols,
    float* __restrict__ out, int E)
{
    const int e = blockIdx.x;
    if (e >= E) return;
    const int   r = rows[e];
    const int   c = cols[e];
    const float v = vals[e];
    const int   t = threadIdx.x;                 // 0..255 -> 2 channels each

    const float2 m = ((const float2*)(S + (size_t)c * OUT_CH))[t];
    float* o = out + (size_t)r * OUT_CH + 2 * t;
    __hip_atomic_fetch_add(o,     v * m.x, __ATOMIC_RELAXED, __HIP_MEMORY_SCOPE_AGENT);
    __hip_atomic_fetch_add(o + 1, v * m.y, __ATOMIC_RELAXED, __HIP_MEMORY_SCOPE_AGENT);
}

// ---------------------------------------------------------------------------
// Kernel 0: zero the (poisoned) output before atomic accumulation
// ---------------------------------------------------------------------------
__global__ void gcn_zero_f4(float4* __restrict__ p, int n4)
{
    const int i = blockIdx.x * blockDim.x + threadIdx.x;
    if (i < n4) p[i] = make_float4(0.f, 0.f, 0.f, 0.f);
}

extern "C" void kernel_launch(void* const* d_in, const int* in_sizes, int n_in,
                              void* d_out, int out_size, void* d_ws, size_t ws_size,
                              hipStream_t stream)
{
    const float* x    = (const float*)d_in[0];   // [M, 512]
    const float* vals = (const float*)d_in[1];   // [E]
    const int*   rows = (const int*)  d_in[2];   // [E]
    const int*   cols = (const int*)  d_in[3];   // [E]
    const float* w    = (const float*)d_in[4];   // [512, 512]
    float*       out  = (float*)d_out;           // [M, 512]
    float*       supp = (float*)d_ws;            // scratch: M*512 floats (102.4 MB)

    const int M = in_sizes[0] / IN_CH;
    const int E = in_sizes[1];

    // 1) out = 0
    const int n4 = out_size / 4;
    gcn_zero_f4<<<(n4 + 255) / 256, 256, 0, stream>>>((float4*)out, n4);

    // 2) supp = x @ W  (fp32 WMMA)
    dim3 grid((M + 127) / 128, OUT_CH / 64);
    gcn_gemm_wmma_f32<<<grid, 128, 0, stream>>>(x, w, supp, M);

    // 3) out[row] += val * supp[col]
    gcn_spmm_scatter<<<E, 256, 0, stream>>>(supp, vals, rows, cols, out, E);
}